// SiamesePlus_6416681141058
// MI455X (gfx1250) — compile-verified
//
#include <hip/hip_runtime.h>
#include <hip/hip_bf16.h>
#include <stdint.h>

// ---------------------------------------------------------------------------
// Siamese PointNet++ for MI455X (gfx1250, wave32, WMMA f32 16x16x4).
// GEMMs: one wave per 16x64 output strip (4 accumulators, A-fragment reuse),
// W strip staged per-block into LDS via async global->LDS when available.
// ---------------------------------------------------------------------------

typedef __attribute__((ext_vector_type(2))) float v2f;
typedef __attribute__((ext_vector_type(8))) float v8f;

typedef __attribute__((address_space(1))) int as1_int;
typedef __attribute__((address_space(3))) int as3_int;

#define LRELU_SLOPE 0.01f
#define BN_EPS 1e-5f

#if defined(__has_builtin)
#if __has_builtin(__builtin_amdgcn_global_load_async_to_lds_b32)
#define HAVE_ASYNC_LDS 1
#endif
#if __has_builtin(__builtin_amdgcn_s_wait_asynccnt)
#define HAVE_WAIT_ASYNC_BUILTIN 1
#endif
#endif
#ifndef HAVE_ASYNC_LDS
#define HAVE_ASYNC_LDS 0
#endif
#ifndef HAVE_WAIT_ASYNC_BUILTIN
#define HAVE_WAIT_ASYNC_BUILTIN 0
#endif

__device__ __forceinline__ void wait_async_lds()
{
#if HAVE_ASYNC_LDS
#if HAVE_WAIT_ASYNC_BUILTIN
    __builtin_amdgcn_s_wait_asynccnt(0);
#else
    asm volatile("s_wait_asynccnt 0" ::: "memory");
#endif
#endif
}

// ---------------------------------------------------------------------------
// GEMM: Y[M,C] = X[M,K] * W[C,K]^T + bias.
// Block = 256 threads = 8 waves. Block owns a 64-wide output strip (tn0) and
// 8 consecutive 16-row tiles. If useLds, the 64xK W strip is staged to LDS
// once (async global->LDS) and shared by all 8 waves.
// WMMA f32 16x16x4 layouts (ISA 7.12.2):
//   A 16x4: lanes 0-15 rows, vgpr0/1 = K{0,1}; lanes 16-31 = K{2,3}
//   B 4x16: symmetric with cols; C/D: vgpr v -> (M = v + 8*hi, N = l15).
// Out-of-range rows/cols are CLAMPED on load (rows/cols of D are independent)
// and only stores are guarded, keeping the hot loop free of EXEC churn.
// ---------------------------------------------------------------------------
__global__ void k_gemm_bias(const float* __restrict__ X, const float* __restrict__ W,
                            const float* __restrict__ Bv, float* __restrict__ Y,
                            int M, int K, int C, int useLds)
{
    extern __shared__ float sW[];
    const int lane = threadIdx.x & 31;
    const int wave = threadIdx.x >> 5;
    const int l15  = lane & 15;
    const int hi   = lane >> 4;
    const int tilesM  = (M + 15) >> 4;
    const int tilesM8 = (tilesM + 7) >> 3;
    const int tn0  = blockIdx.x / tilesM8;          // 64-wide strip index
    const int tm   = (blockIdx.x % tilesM8) * 8 + wave;
    const int pitch = ((K + 1) & ~1) + 4;           // even, >=K, bank-spread

    if (useLds) {                                   // uniform branch
        for (int cl = wave; cl < 64; cl += 8) {
            int cg = tn0 * 64 + cl;
            cg = cg < C ? cg : C - 1;
            const float* src = W + (size_t)cg * K;
            float* dst = sW + cl * pitch;
            for (int k = lane; k < K; k += 32) {
#if HAVE_ASYNC_LDS
                __builtin_amdgcn_global_load_async_to_lds_b32(
                    (as1_int*)(src + k), (as3_int*)(dst + k), 0, 0);
#else
                dst[k] = src[k];
#endif
            }
        }
        wait_async_lds();
    }
    __syncthreads();

    const int row  = tm * 16 + l15;
    const int arow = row < M ? row : M - 1;         // clamp (rows independent)
    const float* Xr = X + (size_t)arow * K;

    int ccol[4];
#pragma unroll
    for (int j = 0; j < 4; ++j) {
        const int c = tn0 * 64 + j * 16 + l15;
        ccol[j] = c < C ? c : C - 1;                // clamp (cols independent)
    }

    v8f acc[4];
#pragma unroll
    for (int j = 0; j < 4; ++j)
        acc[j] = (v8f){0.f, 0.f, 0.f, 0.f, 0.f, 0.f, 0.f, 0.f};

    // Main loop: K4 is a multiple of 4; for every call K is even whenever
    // K4 > 0 (K in {64,128,512,1792}), so v2f (b64) accesses stay aligned.
    const int K4 = K & ~3;
    if (useLds) {
        const float* sB0 = sW + (0 * 16 + l15) * pitch;
        const float* sB1 = sW + (1 * 16 + l15) * pitch;
        const float* sB2 = sW + (2 * 16 + l15) * pitch;
        const float* sB3 = sW + (3 * 16 + l15) * pitch;
        for (int k0 = 0; k0 < K4; k0 += 4) {
            const int k2 = k0 + 2 * hi;
            const v2f a  = *(const v2f*)(Xr + k2);
            const v2f b0 = *(const v2f*)(sB0 + k2);
            const v2f b1 = *(const v2f*)(sB1 + k2);
            const v2f b2 = *(const v2f*)(sB2 + k2);
            const v2f b3 = *(const v2f*)(sB3 + k2);
            acc[0] = __builtin_amdgcn_wmma_f32_16x16x4_f32(false, a, false, b0, (short)0, acc[0], false, false);
            acc[1] = __builtin_amdgcn_wmma_f32_16x16x4_f32(false, a, false, b1, (short)0, acc[1], false, false);
            acc[2] = __builtin_amdgcn_wmma_f32_16x16x4_f32(false, a, false, b2, (short)0, acc[2], false, false);
            acc[3] = __builtin_amdgcn_wmma_f32_16x16x4_f32(false, a, false, b3, (short)0, acc[3], false, false);
        }
    } else {
        const float* W0 = W + (size_t)ccol[0] * K;
        const float* W1 = W + (size_t)ccol[1] * K;
        const float* W2 = W + (size_t)ccol[2] * K;
        const float* W3 = W + (size_t)ccol[3] * K;
        for (int k0 = 0; k0 < K4; k0 += 4) {
            const int k2 = k0 + 2 * hi;
            const v2f a  = *(const v2f*)(Xr + k2);
            const v2f b0 = *(const v2f*)(W0 + k2);
            const v2f b1 = *(const v2f*)(W1 + k2);
            const v2f b2 = *(const v2f*)(W2 + k2);
            const v2f b3 = *(const v2f*)(W3 + k2);
            acc[0] = __builtin_amdgcn_wmma_f32_16x16x4_f32(false, a, false, b0, (short)0, acc[0], false, false);
            acc[1] = __builtin_amdgcn_wmma_f32_16x16x4_f32(false, a, false, b1, (short)0, acc[1], false, false);
            acc[2] = __builtin_amdgcn_wmma_f32_16x16x4_f32(false, a, false, b2, (short)0, acc[2], false, false);
            acc[3] = __builtin_amdgcn_wmma_f32_16x16x4_f32(false, a, false, b3, (short)0, acc[3], false, false);
        }
    }

    if (K4 < K) {                                   // remainder step (K == 3)
        const int k2 = K4 + 2 * hi;
        v2f a;
        a.x = (k2     < K) ? Xr[k2]     : 0.f;
        a.y = (k2 + 1 < K) ? Xr[k2 + 1] : 0.f;
#pragma unroll
        for (int j = 0; j < 4; ++j) {
            const float* base = useLds ? (sW + (j * 16 + l15) * pitch)
                                       : (W + (size_t)ccol[j] * K);
            v2f b;
            b.x = (k2     < K) ? base[k2]     : 0.f;
            b.y = (k2 + 1 < K) ? base[k2 + 1] : 0.f;
            acc[j] = __builtin_amdgcn_wmma_f32_16x16x4_f32(false, a, false, b,
                                                           (short)0, acc[j], false, false);
        }
    }

#pragma unroll
    for (int j = 0; j < 4; ++j) {
        const int col = tn0 * 64 + j * 16 + l15;
        if (col < C) {
            const float bias = Bv[col];
#pragma unroll
            for (int v = 0; v < 8; ++v) {
                const int r = tm * 16 + v + hi * 8;
                if (r < M) Y[(size_t)r * C + col] = acc[j][v] + bias;
            }
        }
    }
}

// ---------------------------------------------------------------------------
// Per-channel mean/var over M rows. One block per channel.
// ---------------------------------------------------------------------------
__global__ void k_bn_stats(const float* __restrict__ Y, float* __restrict__ stats,
                           int M, int C)
{
    const int c = blockIdx.x;
    __shared__ float ss[256];
    __shared__ float sq[256];
    float s = 0.f, q = 0.f;
    for (int r = threadIdx.x; r < M; r += 256) {
        const float v = Y[(size_t)r * C + c];
        s += v; q += v * v;
    }
    ss[threadIdx.x] = s; sq[threadIdx.x] = q;
    __syncthreads();
    for (int off = 128; off > 0; off >>= 1) {
        if ((int)threadIdx.x < off) {
            ss[threadIdx.x] += ss[threadIdx.x + off];
            sq[threadIdx.x] += sq[threadIdx.x + off];
        }
        __syncthreads();
    }
    if (threadIdx.x == 0) {
        const float inv  = 1.f / (float)M;
        const float mean = ss[0] * inv;
        stats[2 * c]     = mean;
        stats[2 * c + 1] = sq[0] * inv - mean * mean;
    }
}

// ---------------------------------------------------------------------------
// In-place BN + (leaky) ReLU.
// ---------------------------------------------------------------------------
__global__ void k_bn_act(float* __restrict__ Y, const float* __restrict__ stats,
                         const float* __restrict__ g, const float* __restrict__ bt,
                         int M, int C, float slope)
{
    const int i = blockIdx.x * blockDim.x + threadIdx.x;
    if (i >= M * C) return;
    const int c = i % C;
    const float mean = stats[2 * c];
    const float var  = stats[2 * c + 1];
    float x = (Y[i] - mean) * rsqrtf(var + BN_EPS) * g[c] + bt[c];
    Y[i] = (x > 0.f) ? x : slope * x;
}

// ---------------------------------------------------------------------------
// Per-batch per-channel max. grid = B * C ; out[b*outStride + outOff + c]
// ---------------------------------------------------------------------------
__global__ void k_batch_colmax(const float* __restrict__ Y, float* __restrict__ out,
                               int rowsPerBatch, int C, int outStride, int outOff)
{
    const int b = blockIdx.x / C;
    const int c = blockIdx.x % C;
    __shared__ float sm[256];
    const float* base = Y + (size_t)b * rowsPerBatch * C;
    float m = -3.402823466e38f;
    for (int r = threadIdx.x; r < rowsPerBatch; r += 256)
        m = fmaxf(m, base[(size_t)r * C + c]);
    sm[threadIdx.x] = m;
    __syncthreads();
    for (int off = 128; off > 0; off >>= 1) {
        if ((int)threadIdx.x < off)
            sm[threadIdx.x] = fmaxf(sm[threadIdx.x], sm[threadIdx.x + off]);
        __syncthreads();
    }
    if (threadIdx.x == 0) out[b * outStride + outOff + c] = sm[0];
}

// ---------------------------------------------------------------------------
// Farthest point sampling. One block per batch; dist[] in LDS.
// ---------------------------------------------------------------------------
__global__ void k_fps(const float* __restrict__ xyz, int* __restrict__ fidx,
                      float* __restrict__ new_xyz, int N, int S)
{
    const int b = blockIdx.x;
    const float* pts = xyz + (size_t)b * N * 3;
    __shared__ float dist[2048];
    __shared__ float rmax[256];
    __shared__ int   rarg[256];
    __shared__ int   s_far;
    for (int i = threadIdx.x; i < N; i += 256) dist[i] = 1e10f;
    if (threadIdx.x == 0) s_far = 0;
    __syncthreads();
    for (int s = 0; s < S; ++s) {
        const int far = s_far;
        const float cx = pts[far * 3 + 0];
        const float cy = pts[far * 3 + 1];
        const float cz = pts[far * 3 + 2];
        if (threadIdx.x == 0) {
            fidx[b * S + s] = far;
            new_xyz[((size_t)b * S + s) * 3 + 0] = cx;
            new_xyz[((size_t)b * S + s) * 3 + 1] = cy;
            new_xyz[((size_t)b * S + s) * 3 + 2] = cz;
        }
        float bm = -1.f; int ba = 0;
        for (int i = threadIdx.x; i < N; i += 256) {
            const float dx = pts[i * 3 + 0] - cx;
            const float dy = pts[i * 3 + 1] - cy;
            const float dz = pts[i * 3 + 2] - cz;
            const float nd = fminf(dist[i], dx * dx + dy * dy + dz * dz);
            dist[i] = nd;
            if (nd > bm) { bm = nd; ba = i; }
        }
        rmax[threadIdx.x] = bm; rarg[threadIdx.x] = ba;
        __syncthreads();
        for (int off = 128; off > 0; off >>= 1) {
            if ((int)threadIdx.x < off) {
                const float om = rmax[threadIdx.x + off];
                const int   oa = rarg[threadIdx.x + off];
                if (om > rmax[threadIdx.x] ||
                    (om == rmax[threadIdx.x] && oa < rarg[threadIdx.x])) {
                    rmax[threadIdx.x] = om; rarg[threadIdx.x] = oa;
                }
            }
            __syncthreads();
        }
        if (threadIdx.x == 0) s_far = rarg[0];
        __syncthreads();
    }
}

// ---------------------------------------------------------------------------
// Ball query + grouping: first K ascending indices with d<=r2, padded with
// first hit; centered coords out. One wave32 per (b,s).
// ---------------------------------------------------------------------------
__global__ void k_group(const float* __restrict__ xyz, const float* __restrict__ new_xyz,
                        float* __restrict__ Gin, int N, int S, int K, float r2)
{
    const int b = blockIdx.x / S;
    const int s = blockIdx.x % S;
    const int lane = threadIdx.x;          // blockDim == 32
    const float* pts = xyz + (size_t)b * N * 3;
    const float qx = new_xyz[((size_t)b * S + s) * 3 + 0];
    const float qy = new_xyz[((size_t)b * S + s) * 3 + 1];
    const float qz = new_xyz[((size_t)b * S + s) * 3 + 2];

    int myidx = -1;
    int cnt = 0;                           // uniform across the wave
    for (int base = 0; base < N && cnt < K; base += 32) {
        const int i = base + lane;
        const float dx = pts[i * 3 + 0] - qx;
        const float dy = pts[i * 3 + 1] - qy;
        const float dz = pts[i * 3 + 2] - qz;
        const bool inb = (dx * dx + dy * dy + dz * dz) <= r2;
        unsigned long long mask = __ballot(inb);
        while (mask && cnt < K) {
            const int bit = __ffsll(mask) - 1;
            mask &= mask - 1;
            if (lane == cnt) myidx = base + bit;
            ++cnt;
        }
    }
    const int first = __shfl(myidx, 0);
    if (myidx < 0) myidx = first;
    if (lane < K) {
        const float* p = pts + (size_t)myidx * 3;
        const size_t row = ((size_t)(b * S + s)) * K + lane;
        Gin[row * 3 + 0] = p[0] - qx;
        Gin[row * 3 + 1] = p[1] - qy;
        Gin[row * 3 + 2] = p[2] - qz;
    }
}

// ---------------------------------------------------------------------------
// Concat [g1(512) | l1(384) | g2(512) | l2(384)] -> h (8,1792)
// ---------------------------------------------------------------------------
__global__ void k_concat(const float* __restrict__ g1, const float* __restrict__ l1,
                         const float* __restrict__ g2, const float* __restrict__ l2,
                         float* __restrict__ h)
{
    const int i = blockIdx.x * blockDim.x + threadIdx.x;
    if (i >= 8 * 1792) return;
    const int b = i / 1792;
    const int c = i % 1792;
    float v;
    if      (c < 512)  v = g1[b * 512 + c];
    else if (c < 896)  v = l1[b * 384 + (c - 512)];
    else if (c < 1408) v = g2[b * 512 + (c - 896)];
    else               v = l2[b * 384 + (c - 1408)];
    h[i] = v;
}

// ---------------------------------------------------------------------------
// Host-side orchestration
// ---------------------------------------------------------------------------
struct Layer { const float *W, *b, *g, *bt; };

static inline void launch_gemm(hipStream_t st, const float* X, const Layer& p,
                               float* Y, int M, int K, int C)
{
    const int tilesM  = (M + 15) / 16;
    const int tilesM8 = (tilesM + 7) / 8;
    const int strips  = (C + 63) / 64;
    const int pitch   = ((K + 1) & ~1) + 4;
    const size_t ldsBytes = (size_t)64 * pitch * sizeof(float);
    const int useLds = (C % 64 == 0) && (ldsBytes <= 160u * 1024u);
    k_gemm_bias<<<dim3(tilesM8 * strips), dim3(256),
                  useLds ? ldsBytes : 0, st>>>(X, p.W, p.b, Y, M, K, C, useLds);
}

static inline void launch_bn_act(hipStream_t st, float* Y, float* stats,
                                 const Layer& p, int M, int C, float slope)
{
    k_bn_stats<<<dim3(C), dim3(256), 0, st>>>(Y, stats, M, C);
    const int n = M * C;
    k_bn_act<<<dim3((n + 255) / 256), dim3(256), 0, st>>>(Y, stats, p.g, p.bt,
                                                          M, C, slope);
}

extern "C" void kernel_launch(void* const* d_in, const int* in_sizes, int n_in,
                              void* d_out, int out_size, void* d_ws, size_t ws_size,
                              hipStream_t stream)
{
    (void)in_sizes; (void)n_in; (void)out_size; (void)ws_size;
    const int B = 8, N = 2048, S = 1024;

    const float* x1 = (const float*)d_in[0];
    const float* x2 = (const float*)d_in[1];
    Layer pf[4], sa[3][3], dec[2];
    int idx = 2;
    for (int l = 0; l < 4; ++l) {
        pf[l] = { (const float*)d_in[idx], (const float*)d_in[idx + 1],
                  (const float*)d_in[idx + 2], (const float*)d_in[idx + 3] };
        idx += 4;
    }
    for (int r = 0; r < 3; ++r)
        for (int l = 0; l < 3; ++l) {
            sa[r][l] = { (const float*)d_in[idx], (const float*)d_in[idx + 1],
                         (const float*)d_in[idx + 2], (const float*)d_in[idx + 3] };
            idx += 4;
        }
    for (int l = 0; l < 2; ++l) {
        dec[l] = { (const float*)d_in[idx], (const float*)d_in[idx + 1],
                   (const float*)d_in[idx + 2], (const float*)d_in[idx + 3] };
        idx += 4;
    }

    // workspace carve-out
    char* ws = (char*)d_ws;
    size_t off = 0;
    auto carve = [&](size_t bytes) -> char* {
        char* p = ws + off;
        off = (off + bytes + 255) & ~(size_t)255;
        return p;
    };
    float* bufA  = (float*)carve((size_t)B * N * 512 * 4);
    float* bufB  = (float*)carve((size_t)B * N * 512 * 4);
    float* stats = (float*)carve(2 * 512 * 4);
    float* gfe0  = (float*)carve((size_t)B * 512 * 4);
    float* gfe1  = (float*)carve((size_t)B * 512 * 4);
    float* lfe0  = (float*)carve((size_t)B * 384 * 4);
    float* lfe1  = (float*)carve((size_t)B * 384 * 4);
    int*   fidx  = (int*)  carve((size_t)B * S * 4);
    float* nxyz  = (float*)carve((size_t)B * S * 3 * 4);
    float* Gin   = (float*)carve((size_t)B * S * 32 * 3 * 4);
    float* Gb1   = (float*)carve((size_t)B * S * 32 * 128 * 4);
    float* Gb2   = (float*)carve((size_t)B * S * 32 * 64 * 4);
    float* hbuf  = (float*)carve((size_t)B * 1792 * 4);
    float* h1    = (float*)carve((size_t)B * 512 * 4);

    const int   pfC[5]    = {3, 64, 128, 512, 512};
    const int   Ks[3]     = {8, 16, 32};
    const float radii2[3] = {0.01f, 0.04f, 0.16f};
    float* gfe[2] = {gfe0, gfe1};
    float* lfe[2] = {lfe0, lfe1};

    for (int br = 0; br < 2; ++br) {
        const float* x = br ? x2 : x1;
        const int M = B * N;

        // ---- pointnet1d : 3 -> 64 -> 128 -> 512 -> 512, BN + leaky ReLU ----
        const float* cur = x;
        float* nxt = bufA;
        float* other = bufB;
        for (int l = 0; l < 4; ++l) {
            launch_gemm(stream, cur, pf[l], nxt, M, pfC[l], pfC[l + 1]);
            launch_bn_act(stream, nxt, stats, pf[l], M, pfC[l + 1], LRELU_SLOPE);
            cur = nxt;
            float* t = nxt; nxt = other; other = t;
        }
        k_batch_colmax<<<dim3(B * 512), dim3(256), 0, stream>>>(
            cur, gfe[br], N, 512, 512, 0);

        // ---- set abstraction ----
        k_fps<<<dim3(B), dim3(256), 0, stream>>>(x, fidx, nxyz, N, S);
        for (int r = 0; r < 3; ++r) {
            const int K = Ks[r];
            k_group<<<dim3(B * S), dim3(32), 0, stream>>>(x, nxyz, Gin,
                                                          N, S, K, radii2[r]);
            const int rows = B * S * K;
            launch_gemm(stream, Gin, sa[r][0], Gb1, rows, 3, 64);
            launch_bn_act(stream, Gb1, stats, sa[r][0], rows, 64, 0.f);
            launch_gemm(stream, Gb1, sa[r][1], Gb2, rows, 64, 64);
            launch_bn_act(stream, Gb2, stats, sa[r][1], rows, 64, 0.f);
            launch_gemm(stream, Gb2, sa[r][2], Gb1, rows, 64, 128);
            launch_bn_act(stream, Gb1, stats, sa[r][2], rows, 128, 0.f);
            k_batch_colmax<<<dim3(B * 128), dim3(256), 0, stream>>>(
                Gb1, lfe[br], S * K, 128, 384, r * 128);
        }
    }

    // ---- decoder ----
    k_concat<<<dim3((8 * 1792 + 255) / 256), dim3(256), 0, stream>>>(
        gfe[0], lfe[0], gfe[1], lfe[1], hbuf);
    launch_gemm(stream, hbuf, dec[0], h1, B, 1792, 512);
    launch_bn_act(stream, h1, stats, dec[0], B, 512, 0.f);
    float* out = (float*)d_out;
    launch_gemm(stream, h1, dec[1], out, B, 512, 1);
    launch_bn_act(stream, out, stats, dec[1], B, 1, 0.f);
}